// UpsampleChunked_83537113907790
// MI455X (gfx1250) — compile-verified
//
#include <hip/hip_runtime.h>
#include <stdint.h>

// ---------------------------------------------------------------------------
// Trilinear 2x upsample, align_corners=True.
// in : (1, 8, 128, 128, 128) f32
// out: (1, 8, 256, 256, 256) f32
//
// Memory-bound (AI ~ 3.8 flop/B, >= 604 MB traffic -> ~26 us floor at
// 23.3 TB/s). Strategy: stage the 4 source rows each output row needs into
// LDS via CDNA5 async global->LDS b128 DMA (ASYNCcnt), compute the separable
// interpolation from LDS, stream results with non-temporal b64 stores so the
// 192MB L2 stays dedicated to the 64MiB input (read from HBM exactly once
// despite 4-9x cross-block reuse).
// ---------------------------------------------------------------------------

#define GLOBAL_AS __attribute__((address_space(1)))
#define LDS_AS    __attribute__((address_space(3)))

#if defined(__AMDGCN__) && __has_builtin(__builtin_amdgcn_global_load_async_to_lds_b128)
#define USE_ASYNC_LDS 1
#else
#define USE_ASYNC_LDS 0
#endif

typedef float f2_t __attribute__((ext_vector_type(2)));
typedef int   v4i  __attribute__((__vector_size__(4 * sizeof(int))));

__device__ __forceinline__ void wait_async_zero() {
#if __has_builtin(__builtin_amdgcn_s_wait_asynccnt)
    __builtin_amdgcn_s_wait_asynccnt(0);
#else
    asm volatile("s_wait_asynccnt 0" ::: "memory");
#endif
}

__global__ __launch_bounds__(128)
void upsample_trilinear2x_kernel(const float* __restrict__ x,
                                 float* __restrict__ out)
{
    // 4 source rows: (d0,h0), (d0,h1), (d1,h0), (d1,h1), 128 f32 each.
    __shared__ float lds[4][128];

    const int t    = threadIdx.x;          // 0..127
    const unsigned bid = blockIdx.x;       // 8*256*256 blocks
    const int ho   = bid & 255;            // output h
    const int dout = (bid >> 8) & 255;     // output d
    const int c    = bid >> 16;            // channel 0..7

    const float R = 0.49803921568627452f;  // (S-1)/(S_out-1) = 127/255

    // ---- d axis weights (uniform per block) ----
    const float cd = (float)dout * R;
    int d0 = (int)cd; if (d0 > 127) d0 = 127;
    int d1 = d0 + 1;  if (d1 > 127) d1 = 127;
    const float fd = cd - (float)d0;

    // ---- h axis weights (uniform per block) ----
    const float ch = (float)ho * R;
    int h0 = (int)ch; if (h0 > 127) h0 = 127;
    int h1 = h0 + 1;  if (h1 > 127) h1 = 127;
    const float fh = ch - (float)h0;

    // ---- stage the 4 input rows into LDS (1 x b128 per thread) ----
    const int row  = t >> 5;               // 0..3 : which source row
    const int seg  = t & 31;               // 16B segment within the row
    const int dsel = (row >> 1) ? d1 : d0;
    const int hsel = (row & 1)  ? h1 : h0;
    const size_t src_off =
        (((size_t)c * 128 + (size_t)dsel) * 128 + (size_t)hsel) * 128
        + (size_t)(seg * 4);

#if USE_ASYNC_LDS
    {
        float* gp = const_cast<float*>(x + src_off);
        __builtin_amdgcn_global_load_async_to_lds_b128(
            (GLOBAL_AS v4i*)gp,
            (LDS_AS v4i*)&lds[row][seg * 4],
            /*imm offset*/ 0, /*cpol*/ 0);
        wait_async_zero();                 // drain this wave's ASYNCcnt
    }
#else
    {
        const float4 v = *(const float4*)(x + src_off);
        *(float4*)&lds[row][seg * 4] = v;
    }
#endif
    __syncthreads();

    const float gd0 = 1.0f - fd;
    const float gh0 = 1.0f - fh;

    // ---- each thread produces 2 contiguous outputs along W ----
    float res[2];
#pragma unroll
    for (int j = 0; j < 2; ++j) {
        const int wo = 2 * t + j;
        const float cw = (float)wo * R;
        int w0 = (int)cw; if (w0 > 127) w0 = 127;
        int w1 = w0 + 1;  if (w1 > 127) w1 = 127;
        const float fw  = cw - (float)w0;
        const float gw0 = 1.0f - fw;

        const float v00 = lds[0][w0] * gw0 + lds[0][w1] * fw; // (d0,h0)
        const float v01 = lds[1][w0] * gw0 + lds[1][w1] * fw; // (d0,h1)
        const float v10 = lds[2][w0] * gw0 + lds[2][w1] * fw; // (d1,h0)
        const float v11 = lds[3][w0] * gw0 + lds[3][w1] * fw; // (d1,h1)

        const float a = v00 * gh0 + v01 * fh;
        const float b = v10 * gh0 + v11 * fh;
        res[j] = a * gd0 + b * fd;
    }

    // ---- streaming (non-temporal) b64 store, write-once output ----
    const size_t dst_off =
        (((size_t)c * 256 + (size_t)dout) * 256 + (size_t)ho) * 256
        + (size_t)(2 * t);
    f2_t o2; o2.x = res[0]; o2.y = res[1];
    __builtin_nontemporal_store(o2, (f2_t*)(out + dst_off));
}

extern "C" void kernel_launch(void* const* d_in, const int* in_sizes, int n_in,
                              void* d_out, int out_size, void* d_ws, size_t ws_size,
                              hipStream_t stream)
{
    (void)in_sizes; (void)n_in; (void)out_size; (void)d_ws; (void)ws_size;

    const float* x  = (const float*)d_in[0];   // (1,8,128,128,128) f32
    float*      out = (float*)d_out;           // (1,8,256,256,256) f32

    // one block per output (c, d_out, h_out) row: 8*256*256 blocks, 128 thr
    dim3 grid(8u * 256u * 256u);
    dim3 block(128);
    upsample_trilinear2x_kernel<<<grid, block, 0, stream>>>(x, out);
}